// My_super_loss_groupouter_52725018526339
// MI455X (gfx1250) — compile-verified
//
#include <hip/hip_runtime.h>
#include <cmath>

// MI455X / gfx1250, wave32. Workload is latency-bound (FPS is 511 serial
// argmax steps); all data (33.6MB) is L2-resident (192MB L2). The KNN
// dot-product matmul (K=3 padded to 4) runs on V_WMMA_F32_16X16X4_F32,
// 4 WMMAs per iteration (64 support points) to overlap issue and cut
// barrier count; top-32 selection uses all 32 lanes (2 lanes/query).

typedef __attribute__((ext_vector_type(2))) float v2f;
typedef __attribute__((ext_vector_type(8))) float v8f;

static constexpr int kB   = 8;      // batches
static constexpr int kN   = 8192;   // points
static constexpr int kC   = 64;     // feature dims
static constexpr int kDS  = 512;    // fps samples (N/16)
static constexpr int kK   = 32;     // knn neighbors
static constexpr int kKK  = 33;     // group rows (k neighbors + center)
#define COS_EPS 1e-8f
#define TCR_SCALAR 40.0f            // c/(BS*eps) = 64/(8*0.2)

// ---------------------------------------------------------------- init
__global__ void k_init(float* acc, float* S) {
  if (threadIdx.x == 0) acc[0] = 0.0f;
  if (threadIdx.x < kB) S[threadIdx.x] = 0.0f;
}

// ------------------------------------------- global per-point cosine loss
__global__ void k_global(const float* __restrict__ a,
                         const float* __restrict__ b,
                         float* __restrict__ acc) {
  int pt = blockIdx.x * blockDim.x + threadIdx.x;   // 0..B*N-1
  const float* pa = a + (size_t)pt * kC;
  const float* pb = b + (size_t)pt * kC;
  float dot = 0.f, na = 0.f, nb = 0.f;
  #pragma unroll
  for (int c = 0; c < kC; c += 4) {
    float4 x = *(const float4*)(pa + c);
    float4 y = *(const float4*)(pb + c);
    dot += x.x*y.x + x.y*y.y + x.z*y.z + x.w*y.w;
    na  += x.x*x.x + x.y*x.y + x.z*x.z + x.w*x.w;
    nb  += y.x*y.x + y.y*y.y + y.z*y.z + y.w*y.w;
  }
  float cosv = dot / fmaxf(sqrtf(na) * sqrtf(nb), COS_EPS);
  __shared__ float red[256];
  red[threadIdx.x] = cosv;
  __syncthreads();
  for (int s = 128; s > 0; s >>= 1) {
    if (threadIdx.x < s) red[threadIdx.x] += red[threadIdx.x + s];
    __syncthreads();
  }
  if (threadIdx.x == 0) atomicAdd(acc, red[0]);
}

// ---------------------------------------------------------------- FPS
// One workgroup per (view,batch). dist[] lives in LDS; 511 serial steps of
// (argmax over 8192) -> (min-update). Tie-break: lowest index (jnp.argmax).
__global__ void k_fps(const float* __restrict__ pf,
                      const float* __restrict__ ps,
                      int* __restrict__ idx_out) {
  int view = blockIdx.x >> 3;
  int b    = blockIdx.x & 7;
  const float* p = (view == 0 ? pf : ps) + (size_t)b * kN * 3;
  int* idx = idx_out + ((size_t)view * kB + b) * kDS;

  __shared__ float dist[kN];
  __shared__ float red_d[256];
  __shared__ int   red_i[256];
  __shared__ float cur[3];

  int tid = threadIdx.x;
  float sx = p[0], sy = p[1], sz = p[2];     // seed = point 0
  for (int i = tid; i < kN; i += 256) {
    float dx = p[i*3] - sx, dy = p[i*3+1] - sy, dz = p[i*3+2] - sz;
    dist[i] = dx*dx + dy*dy + dz*dz;
  }
  if (tid == 0) idx[0] = 0;
  __syncthreads();

  for (int step = 1; step < kDS; ++step) {
    float bd = -1.0f; int bi = kN;
    for (int i = tid; i < kN; i += 256) {
      float d = dist[i];
      if (d > bd) { bd = d; bi = i; }        // strict '>' keeps lowest i
    }
    red_d[tid] = bd; red_i[tid] = bi;
    __syncthreads();
    for (int s = 128; s > 0; s >>= 1) {
      if (tid < s) {
        float d2 = red_d[tid + s]; int i2 = red_i[tid + s];
        if (d2 > red_d[tid] || (d2 == red_d[tid] && i2 < red_i[tid])) {
          red_d[tid] = d2; red_i[tid] = i2;
        }
      }
      __syncthreads();
    }
    if (tid == 0) {
      int nxt = red_i[0];
      idx[step] = nxt;
      cur[0] = p[nxt*3]; cur[1] = p[nxt*3+1]; cur[2] = p[nxt*3+2];
    }
    __syncthreads();
    float cx = cur[0], cy = cur[1], cz = cur[2];
    for (int i = tid; i < kN; i += 256) {
      float dx = p[i*3] - cx, dy = p[i*3+1] - cy, dz = p[i*3+2] - cz;
      float d = dx*dx + dy*dy + dz*dz;
      if (d < dist[i]) dist[i] = d;
    }
    __syncthreads();
  }
}

// ------------------------------------------- KNN (WMMA f32 16x16x4) + group loss
// One wave (32 lanes) per (b, view, 16-query tile). Per iteration: 64 support
// points via 4 WMMAs; selection on all 32 lanes (lane m: cols 0-31, lane m+16:
// cols 32-63), sorted lists merged stably (tie -> lower global index).
__global__ void k_group(const float* __restrict__ logits,
                        const float* __restrict__ logits1,
                        const float* __restrict__ pf,
                        const float* __restrict__ ps,
                        const int* __restrict__ idx_all,
                        float* __restrict__ S) {
  int qtile = blockIdx.x;   // 0..31
  int view  = blockIdx.y;   // 0..1
  int b     = blockIdx.z;   // 0..7
  const float* p = (view == 0 ? pf : ps) + (size_t)b * kN * 3;
  const float* F = (view == 0 ? logits : logits1) + (size_t)b * kN * kC;
  const int* idx = idx_all + ((size_t)view * kB + b) * kDS + qtile * 16;

  __shared__ float qx[16], qy[16], qz[16], qq[16];
  __shared__ int   cidx[16];
  __shared__ float dtile[16][68];        // 16 queries x 64 support cols (+pad)
  __shared__ float td2[32][kK];          // per-lane partial top-32 (sorted)
  __shared__ int   ti2[32][kK];
  __shared__ float top_d[16][kK];        // merged per-query top-32
  __shared__ int   top_i[16][kK];

  int lane = threadIdx.x;       // 0..31
  int m16  = lane & 15;

  if (lane < 16) {
    int gi = idx[lane];
    cidx[lane] = gi;
    float x = p[gi*3], y = p[gi*3+1], z = p[gi*3+2];
    qx[lane] = x; qy[lane] = y; qz[lane] = z;
    qq[lane] = x*x + y*y + z*z;
  }
  for (int j = 0; j < kK; ++j) { td2[lane][j] = 3.4e38f; ti2[lane][j] = 0x7fffffff; }
  __syncthreads();

  // A operand: 16x4 f32 layout — lanes 0-15: {K0=x[M],K1=y[M]}, lanes 16-31: {K2=z[M],K3=0}
  bool lo = (lane < 16);
  v2f A;
  A.x = lo ? qx[m16] : qz[m16];
  A.y = lo ? qy[m16] : 0.0f;
  int mbase = (lane >> 4) * 8;

  for (int sb = 0; sb < kN; sb += 64) {
    // prefetch next 64-point block (global_prefetch_b8)
    if (sb + 64 < kN) __builtin_prefetch(p + (size_t)(sb + 64 + m16) * 3, 0, 0);
    #pragma unroll
    for (int c = 0; c < 4; ++c) {
      int n = sb + c * 16 + m16;
      float x = p[n*3], y = p[n*3+1], z = p[n*3+2];
      float ssv = x*x + y*y + z*z;
      // B operand: 4x16 f32 — lanes 0-15: {K0=x[N],K1=y[N]}, lanes 16-31: {K2=z[N],K3=0}
      v2f Bv;
      Bv.x = lo ? x : z;
      Bv.y = lo ? y : 0.0f;
      v8f C = {};
      C = __builtin_amdgcn_wmma_f32_16x16x4_f32(false, A, false, Bv,
                                                (short)0, C, false, false);
      #pragma unroll
      for (int r = 0; r < 8; ++r) {
        int M = mbase + r;
        dtile[M][c * 16 + m16] = qq[M] + ssv - 2.0f * C[r];
      }
    }
    __syncthreads();
    {
      // lane = (half<<4) | q : query q, columns half*32 .. half*32+31
      int q = m16, base = (lane >> 4) * 32;
      for (int j = 0; j < 32; ++j) {
        float dv = dtile[q][base + j];
        if (dv < td2[lane][kK - 1]) {      // strict '<' keeps stable tie order
          int pos = kK - 1;
          while (pos > 0 && td2[lane][pos - 1] > dv) {
            td2[lane][pos] = td2[lane][pos - 1];
            ti2[lane][pos] = ti2[lane][pos - 1];
            --pos;
          }
          td2[lane][pos] = dv; ti2[lane][pos] = sb + base + j;
        }
      }
    }
    __syncthreads();
  }

  // merge the two sorted partial lists per query (stable: tie -> lower index)
  if (lane < 16) {
    int i = 0, j = 0;
    for (int t = 0; t < kK; ++t) {
      float da = td2[lane][i],      db = td2[lane + 16][j];
      int   ia = ti2[lane][i],      ib = ti2[lane + 16][j];
      bool takeA = (da < db) || (da == db && ia < ib);
      if (takeA) { top_d[lane][t] = da; top_i[lane][t] = ia; ++i; }
      else       { top_d[lane][t] = db; top_i[lane][t] = ib; ++j; }
    }
  }
  __syncthreads();

  // ---- group losses: D = [32 neighbor rows ; center row], c=64
  // lane owns dims {lane, lane+32}; butterfly shuffles for reductions.
  float gsum = 0.0f;
  for (int m = 0; m < 16; ++m) {
    float a0 = 0.f, a1 = 0.f;
    for (int r = 0; r < kKK; ++r) {
      int ri = (r < kK) ? top_i[m][r] : cidx[m];
      const float* fr = F + (size_t)ri * kC;
      a0 += fr[lane]; a1 += fr[lane + 32];
    }
    a0 *= (1.0f / kKK); a1 *= (1.0f / kKK);

    float an = a0*a0 + a1*a1;
    for (int o = 16; o > 0; o >>= 1) an += __shfl_xor(an, o, 32);
    float avg_norm = sqrtf(an);

    float csum = 0.f, tsum = 0.f;
    for (int r = 0; r < kKK; ++r) {
      int ri = (r < kK) ? top_i[m][r] : cidx[m];
      const float* fr = F + (size_t)ri * kC;
      float f0 = fr[lane], f1 = fr[lane + 32];
      float dd = f0*a0 + f1*a1;
      float nn = f0*f0 + f1*f1;
      for (int o = 16; o > 0; o >>= 1) {
        dd += __shfl_xor(dd, o, 32);
        nn += __shfl_xor(nn, o, 32);
      }
      csum += dd / fmaxf(sqrtf(nn) * avg_norm, COS_EPS);
      tsum += -0.5f * log1pf(TCR_SCALAR * nn);
    }
    float loss_contract = -(csum / kKK);
    float loss_tcr      = tsum / kKK;
    gsum += 200.0f * loss_contract + loss_tcr;
  }
  if (lane == 0) atomicAdd(&S[b], gsum);
}

// ---------------------------------------------------------------- finalize
// Faithful sequential fold: g = (g + S[b]) / ds, then /b; out = gp + 2g.
__global__ void k_final(const float* __restrict__ acc,
                        const float* __restrict__ S,
                        float* __restrict__ out) {
  if (threadIdx.x == 0 && blockIdx.x == 0) {
    float g = 0.0f;
    for (int b = 0; b < kB; ++b) g = (g + S[b]) / (float)kDS;
    g /= (float)kB;
    float gp = -acc[0] / (float)(kB * kN);
    out[0] = gp + 2.0f * g;
  }
}

extern "C" void kernel_launch(void* const* d_in, const int* in_sizes, int n_in,
                              void* d_out, int out_size, void* d_ws, size_t ws_size,
                              hipStream_t stream) {
  const float* logits  = (const float*)d_in[0];
  const float* logits1 = (const float*)d_in[1];
  const float* pf      = (const float*)d_in[2];
  const float* ps      = (const float*)d_in[3];
  // d_in[4] (orixyz) is unused by the reference's returned loss.

  float* wsf = (float*)d_ws;
  float* acc = wsf;               // [1]  global cosine accumulator
  float* S   = wsf + 8;           // [8]  per-batch group sums
  int*   idx = (int*)(wsf + 16);  // [2][8][512] fps indices (32KB)
  float* out = (float*)d_out;

  k_init<<<1, 32, 0, stream>>>(acc, S);
  k_global<<<(kB * kN) / 256, 256, 0, stream>>>(logits, logits1, acc);
  k_fps<<<16, 256, 0, stream>>>(pf, ps, idx);
  dim3 g3(32, 2, kB);
  k_group<<<g3, 32, 0, stream>>>(logits, logits1, pf, ps, idx, S);
  k_final<<<1, 32, 0, stream>>>(acc, S, out);
}